// OSQP_62448824484655
// MI455X (gfx1250) — compile-verified
//
#include <hip/hip_runtime.h>

typedef __attribute__((ext_vector_type(2))) float v2f;
typedef __attribute__((ext_vector_type(8))) float v8f;

#define NV 256      // variables
#define MC 512      // constraints
#define NB 64       // batch
#define SIGMA_C 1e-6f
#define RHO_C 0.1f
#define ALPHA_C 1.6f
#define N_ITERS 500

// ---------------------------------------------------------------------------
// Kernel 1: Mk = RHO * A^T A + diag(P + SIGMA)   via V_WMMA_F32_16X16X4_F32
// One wave = one 16x16 tile of Mk[b]. K (m) streamed in chunks of 4.
// fp32 WMMA A-layout (16x4): lanes 0-15 hold {K=0,K=1}, lanes 16-31 {K=2,K=3};
// B-layout (4x16) mirrors it. Amat[r][k] = A[m+k][ti*16+r], Bmat[k][c] = A[m+k][tj*16+c].
// ---------------------------------------------------------------------------
__global__ __launch_bounds__(256) void build_kkt(const float* __restrict__ A,
                                                 const float* __restrict__ P,
                                                 float* __restrict__ Mk) {
  const int b    = blockIdx.y;
  const int wave = threadIdx.x >> 5;
  const int lane = threadIdx.x & 31;
  const int half = lane >> 4;
  const int l    = lane & 15;
  const int tile = blockIdx.x * 8 + wave;   // 0..255
  const int ti   = tile >> 4;
  const int tj   = tile & 15;

  const float* Ab = A + (size_t)b * MC * NV;
  const int ca = ti * 16 + l;               // column of A feeding rows of tile
  const int cb = tj * 16 + l;               // column of A feeding cols of tile

  v8f c = {0.f, 0.f, 0.f, 0.f, 0.f, 0.f, 0.f, 0.f};
  for (int m = 0; m < MC; m += 4) {
    const int r0 = m + 2 * half;            // this lane's two K rows
    v2f a, bm;
    a.x  = Ab[(size_t)r0 * NV + ca];
    a.y  = Ab[(size_t)(r0 + 1) * NV + ca];
    bm.x = Ab[(size_t)r0 * NV + cb];
    bm.y = Ab[(size_t)(r0 + 1) * NV + cb];
    c = __builtin_amdgcn_wmma_f32_16x16x4_f32(false, a, false, bm,
                                              (short)0, c, false, false);
  }

  // D layout: lanes 0-15 -> N=l, M=v ; lanes 16-31 -> N=l, M=v+8
  float* out = Mk + (size_t)b * NV * NV;
  #pragma unroll
  for (int v = 0; v < 8; ++v) {
    const int row = ti * 16 + v + 8 * half;
    const int col = cb;
    float val = RHO_C * c[v];
    if (row == col) val += P[b * NV + row] + SIGMA_C;
    out[(size_t)row * NV + col] = val;
  }
}

// ---------------------------------------------------------------------------
// Kernel 2: in-place Gauss-Jordan inverse of SPD Mk[b] (256x256) in LDS.
// Row-wise accesses are bank-conflict-free; column touch is 1 elem/row/pivot.
// 256 threads, one workgroup per batch, 256 KB dynamic LDS.
// ---------------------------------------------------------------------------
extern __shared__ float Sdyn[];

__global__ __launch_bounds__(256) void gj_invert(float* __restrict__ Mk) {
  const int b = blockIdx.x;
  const int j = threadIdx.x;
  float* S = Sdyn;
  float* G = Mk + (size_t)b * NV * NV;

  for (int r = 0; r < NV; ++r) S[r * NV + j] = G[(size_t)r * NV + j];
  __syncthreads();

  for (int p = 0; p < NV; ++p) {
    const float dinv = 1.0f / S[p * NV + p];
    __syncthreads();                          // everyone read pivot
    if (j != p) S[p * NV + j] *= dinv;        // scale pivot row
    __syncthreads();
    const float spj = S[p * NV + j];          // scaled pivot-row element
    for (int i = 0; i < NV; ++i) {            // eliminate all other rows
      if (i == p) continue;
      const float aip = S[i * NV + p];        // broadcast read (same addr)
      if (j != p) S[i * NV + j] -= aip * spj;
    }
    __syncthreads();
    if (j != p) S[j * NV + p] = -S[j * NV + p] * dinv;   // pivot column
    else        S[p * NV + p] = dinv;
    __syncthreads();
  }

  for (int r = 0; r < NV; ++r) G[(size_t)r * NV + j] = S[r * NV + j];
}

// ---------------------------------------------------------------------------
// Kernel 3: persistent ADMM loop. One WG (1024 thr = 32 waves) per batch.
// State vectors live in LDS; A and Minv stream from L2 (48 MB total, resident).
// ---------------------------------------------------------------------------
__global__ __launch_bounds__(1024) void admm_iterate(
    const float* __restrict__ A, const float* __restrict__ Minv,
    const float* __restrict__ q, const float* __restrict__ lvec,
    const float* __restrict__ uvec, float* __restrict__ out) {
  __shared__ float xs[NV], xt[NV], rhs[NV], qs[NV];
  __shared__ float us[MC], zs[MC], ys[MC], zt[MC], lo[MC], hi[MC];
  __shared__ float part[1024];

  const int b    = blockIdx.x;
  const int tid  = threadIdx.x;
  const int lane = tid & 31;
  const int wave = tid >> 5;
  const float* Ab = A    + (size_t)b * MC * NV;
  const float* Mb = Minv + (size_t)b * NV * NV;

  if (tid < NV) { qs[tid] = q[b * NV + tid]; xs[tid] = 0.f; }
  if (tid < MC) {
    lo[tid] = lvec[b * MC + tid];
    hi[tid] = uvec[b * MC + tid];
    zs[tid] = 0.f; ys[tid] = 0.f;
  }
  __syncthreads();

  const int n  = tid & 255;   // output index for GEMV phases
  const int c4 = tid >> 8;    // reduction chunk 0..3

  for (int it = 0; it < N_ITERS; ++it) {
    // u = rho*z - y
    if (tid < MC) us[tid] = RHO_C * zs[tid] - ys[tid];
    __syncthreads();

    // rhs = sigma*x - q + A^T u   (threads along n -> coalesced A reads)
    {
      float acc = 0.f;
      const int m0 = c4 * 128;
      #pragma unroll 4
      for (int m = m0; m < m0 + 128; ++m)
        acc += Ab[(size_t)m * NV + n] * us[m];
      part[tid] = acc;
    }
    __syncthreads();
    if (tid < NV)
      rhs[tid] = SIGMA_C * xs[tid] - qs[tid] +
                 part[tid] + part[tid + 256] + part[tid + 512] + part[tid + 768];
    __syncthreads();

    // x~ = Minv * rhs   (Minv symmetric: read row k instead of column -> coalesced)
    {
      float acc = 0.f;
      const int k0 = c4 * 64;
      #pragma unroll 4
      for (int k = k0; k < k0 + 64; ++k)
        acc += Mb[(size_t)k * NV + n] * rhs[k];
      part[tid] = acc;
    }
    __syncthreads();
    if (tid < NV) {
      const float v = part[tid] + part[tid + 256] + part[tid + 512] + part[tid + 768];
      xt[tid] = v;
      xs[tid] = ALPHA_C * v + (1.f - ALPHA_C) * xs[tid];
    }
    __syncthreads();

    // z~ = A x~  (row dot per wave, wave32 shfl_xor reduction)
    for (int r = 0; r < 16; ++r) {
      const int m = wave * 16 + r;
      float acc = 0.f;
      #pragma unroll
      for (int i = 0; i < 8; ++i) {
        const int col = lane + 32 * i;
        acc += Ab[(size_t)m * NV + col] * xt[col];
      }
      acc += __shfl_xor(acc, 16);
      acc += __shfl_xor(acc, 8);
      acc += __shfl_xor(acc, 4);
      acc += __shfl_xor(acc, 2);
      acc += __shfl_xor(acc, 1);
      if (lane == 0) zt[m] = acc;
    }
    __syncthreads();

    // relaxation + projection + dual update
    if (tid < MC) {
      const float zh = ALPHA_C * zt[tid] + (1.f - ALPHA_C) * zs[tid];
      const float zc = zh + ys[tid] * (1.f / RHO_C);
      const float zn = fminf(fmaxf(zc, lo[tid]), hi[tid]);
      ys[tid] += RHO_C * (zh - zn);
      zs[tid] = zn;
    }
    __syncthreads();
  }

  if (tid < NV) out[b * NV + tid] = xs[tid];
}

// ---------------------------------------------------------------------------
extern "C" void kernel_launch(void* const* d_in, const int* in_sizes, int n_in,
                              void* d_out, int out_size, void* d_ws, size_t ws_size,
                              hipStream_t stream) {
  (void)in_sizes; (void)n_in; (void)out_size; (void)ws_size;
  const float* P = (const float*)d_in[0];
  const float* q = (const float*)d_in[1];
  const float* A = (const float*)d_in[2];
  const float* l = (const float*)d_in[3];
  const float* u = (const float*)d_in[4];
  float* out = (float*)d_out;
  float* Mk  = (float*)d_ws;   // 64 * 256 * 256 floats = 16 MB (Mk, then Minv in place)

  // allow 256 KB dynamic LDS for the inversion kernel (deterministic host call)
  hipFuncSetAttribute(reinterpret_cast<const void*>(gj_invert),
                      hipFuncAttributeMaxDynamicSharedMemorySize,
                      NV * NV * (int)sizeof(float));

  build_kkt<<<dim3(32, NB), 256, 0, stream>>>(A, P, Mk);
  gj_invert<<<NB, NV, NV * NV * sizeof(float), stream>>>(Mk);
  admm_iterate<<<NB, 1024, 0, stream>>>(A, Mk, q, l, u, out);
}